// BipartiteGATConv_1099511628142
// MI455X (gfx1250) — compile-verified
//
#include <hip/hip_runtime.h>
#include <hip/hip_bf16.h>
#include <math.h>

typedef __attribute__((ext_vector_type(2))) float v2f;
typedef __attribute__((ext_vector_type(8))) float v8f;

#define HEADS 4
#define FDIM  128

// ---- order-preserving float <-> uint key (for atomic max on floats) ----
__device__ __forceinline__ unsigned f2key(float f) {
    unsigned b = __float_as_uint(f);
    return (b & 0x80000000u) ? ~b : (b | 0x80000000u);
}
__device__ __forceinline__ float key2f(unsigned k) {
    unsigned b = (k & 0x80000000u) ? (k & 0x7FFFFFFFu) : ~k;
    return __uint_as_float(b);
}

// =====================================================================
// GEMM: Y[M,128] = X[M,128] @ W[128,128] (+bias), via v_wmma_f32_16x16x4_f32
// grid.x = M/16 tiles of rows, 8 waves per block = 8 column tiles of 16
// =====================================================================
__global__ __launch_bounds__(256) void gemm_x128(
    const float* __restrict__ X, const float* __restrict__ W,
    const float* __restrict__ bias, float* __restrict__ Y, int M)
{
    __shared__ float At[16][132];   // pitch 132 dwords -> conflict-free frag reads
    const int tid  = threadIdx.x;
    const int row0 = blockIdx.x * 16;

    // cooperative load of the 16x128 A tile (2 x float4 per thread)
    #pragma unroll
    for (int i = 0; i < 2; ++i) {
        int idx = (tid + i * 256) * 4;          // 0..2047 dwords
        int r = idx >> 7, c = idx & 127;
        int gr = row0 + r; gr = gr < M ? gr : M - 1;
        const float4 v = *(const float4*)(X + (size_t)gr * FDIM + c);
        At[r][c + 0] = v.x; At[r][c + 1] = v.y;
        At[r][c + 2] = v.z; At[r][c + 3] = v.w;
    }
    __syncthreads();

    const int lane = tid & 31;
    const int wv   = tid >> 5;          // column tile 0..7
    const int m    = lane & 15;         // A row / C column index
    const int half = lane >> 4;         // 0/1
    const int koff = half << 1;         // K sub-offset {0,2}
    const int n    = wv * 16 + (lane & 15);

    v8f acc = {};
    #pragma unroll
    for (int kb = 0; kb < FDIM; kb += 4) {
        v2f a, b;
        a.x = At[m][kb + koff];
        a.y = At[m][kb + koff + 1];
        b.x = W[(kb + koff)     * FDIM + n];
        b.y = W[(kb + koff + 1) * FDIM + n];
        acc = __builtin_amdgcn_wmma_f32_16x16x4_f32(
                  false, a, false, b, (short)0, acc, false, false);
    }

    const float bn = bias ? bias[n] : 0.0f;
    #pragma unroll
    for (int r = 0; r < 8; ++r) {       // D layout: VGPR r -> M = r + 8*half
        int row = row0 + half * 8 + r;
        if (row < M) Y[(size_t)row * FDIM + n] = acc[r] + bn;
    }
}

// =====================================================================
// per-node alpha: out[n,h] = sum_d feat[n,h*32+d] * att[h*32+d]
// one wave per node; lane l covers dims [4l, 4l+4) -> head = l/8
// =====================================================================
__global__ __launch_bounds__(256) void node_alpha(
    const float* __restrict__ feat, const float* __restrict__ att,
    float* __restrict__ out, int N)
{
    const int lane = threadIdx.x & 31;
    const int node = blockIdx.x * 8 + (threadIdx.x >> 5);
    if (node >= N) return;
    const float4 f = *(const float4*)(feat + (size_t)node * FDIM + lane * 4);
    const float4 a = *(const float4*)(att + lane * 4);
    float p = f.x * a.x + f.y * a.y + f.z * a.z + f.w * a.w;
    p += __shfl_xor(p, 1);
    p += __shfl_xor(p, 2);
    p += __shfl_xor(p, 4);
    if ((lane & 7) == 0) out[node * HEADS + (lane >> 3)] = p;
}

// =====================================================================
__global__ void init_softmax(unsigned* __restrict__ amax,
                             float* __restrict__ asum, int n)
{
    int i = blockIdx.x * blockDim.x + threadIdx.x;
    if (i < n) { amax[i] = 0u; asum[i] = 0.0f; }
}

__global__ void edge_alpha_max(
    const int* __restrict__ esrc, const int* __restrict__ edst,
    const float* __restrict__ asn, const float* __restrict__ adn,
    float* __restrict__ alpha_e, unsigned* __restrict__ amax, int E)
{
    int e = blockIdx.x * blockDim.x + threadIdx.x;
    if (e >= E) return;
    const int s = esrc[e], d = edst[e];
    const float4 a = *(const float4*)(asn + (size_t)s * HEADS);
    const float4 b = *(const float4*)(adn + (size_t)d * HEADS);
    float4 al;
    al.x = a.x + b.x; al.x = al.x > 0.f ? al.x : 0.2f * al.x;
    al.y = a.y + b.y; al.y = al.y > 0.f ? al.y : 0.2f * al.y;
    al.z = a.z + b.z; al.z = al.z > 0.f ? al.z : 0.2f * al.z;
    al.w = a.w + b.w; al.w = al.w > 0.f ? al.w : 0.2f * al.w;
    *(float4*)(alpha_e + (size_t)e * HEADS) = al;
    unsigned* mp = amax + (size_t)d * HEADS;
    __hip_atomic_fetch_max(mp + 0, f2key(al.x), __ATOMIC_RELAXED, __HIP_MEMORY_SCOPE_AGENT);
    __hip_atomic_fetch_max(mp + 1, f2key(al.y), __ATOMIC_RELAXED, __HIP_MEMORY_SCOPE_AGENT);
    __hip_atomic_fetch_max(mp + 2, f2key(al.z), __ATOMIC_RELAXED, __HIP_MEMORY_SCOPE_AGENT);
    __hip_atomic_fetch_max(mp + 3, f2key(al.w), __ATOMIC_RELAXED, __HIP_MEMORY_SCOPE_AGENT);
}

__global__ void edge_exp_sum(
    const int* __restrict__ edst, const unsigned* __restrict__ amax,
    float* __restrict__ alpha_e, float* __restrict__ asum, int E)
{
    int e = blockIdx.x * blockDim.x + threadIdx.x;
    if (e >= E) return;
    const int d = edst[e];
    float4 al = *(const float4*)(alpha_e + (size_t)e * HEADS);
    const unsigned* mp = amax + (size_t)d * HEADS;
    al.x = __expf(al.x - key2f(mp[0]));
    al.y = __expf(al.y - key2f(mp[1]));
    al.z = __expf(al.z - key2f(mp[2]));
    al.w = __expf(al.w - key2f(mp[3]));
    *(float4*)(alpha_e + (size_t)e * HEADS) = al;
    float* sp = asum + (size_t)d * HEADS;
    __hip_atomic_fetch_add(sp + 0, al.x, __ATOMIC_RELAXED, __HIP_MEMORY_SCOPE_AGENT);
    __hip_atomic_fetch_add(sp + 1, al.y, __ATOMIC_RELAXED, __HIP_MEMORY_SCOPE_AGENT);
    __hip_atomic_fetch_add(sp + 2, al.z, __ATOMIC_RELAXED, __HIP_MEMORY_SCOPE_AGENT);
    __hip_atomic_fetch_add(sp + 3, al.w, __ATOMIC_RELAXED, __HIP_MEMORY_SCOPE_AGENT);
}

// one wave per edge: lane l -> dims [4l,4l+4), head l/8; coalesced 512B gather
__global__ __launch_bounds__(256) void edge_scatter(
    const int* __restrict__ esrc, const int* __restrict__ edst,
    const float* __restrict__ src_feat, const float* __restrict__ alpha_e,
    const float* __restrict__ asum, float* __restrict__ out, int E)
{
    const int lane = threadIdx.x & 31;
    const int e = blockIdx.x * 8 + (threadIdx.x >> 5);
    if (e >= E) return;
    const int s = esrc[e], d = edst[e];
    const int h = lane >> 3;
    const float w = alpha_e[(size_t)e * HEADS + h] /
                    (asum[(size_t)d * HEADS + h] + 1e-16f);
    const float4 f = *(const float4*)(src_feat + (size_t)s * FDIM + lane * 4);
    float* op = out + (size_t)d * FDIM + lane * 4;
    __hip_atomic_fetch_add(op + 0, w * f.x, __ATOMIC_RELAXED, __HIP_MEMORY_SCOPE_AGENT);
    __hip_atomic_fetch_add(op + 1, w * f.y, __ATOMIC_RELAXED, __HIP_MEMORY_SCOPE_AGENT);
    __hip_atomic_fetch_add(op + 2, w * f.z, __ATOMIC_RELAXED, __HIP_MEMORY_SCOPE_AGENT);
    __hip_atomic_fetch_add(op + 3, w * f.w, __ATOMIC_RELAXED, __HIP_MEMORY_SCOPE_AGENT);
}

// =====================================================================
extern "C" void kernel_launch(void* const* d_in, const int* in_sizes, int n_in,
                              void* d_out, int out_size, void* d_ws, size_t ws_size,
                              hipStream_t stream)
{
    const float* x_src  = (const float*)d_in[0];
    const float* x_dst  = (const float*)d_in[1];
    const int*   esrc   = (const int*)d_in[2];
    const int*   edst   = (const int*)d_in[3];
    const float* W_src  = (const float*)d_in[5];
    const float* W_dst  = (const float*)d_in[6];
    const float* attS   = (const float*)d_in[7];
    const float* attD   = (const float*)d_in[8];
    const float* W_self = (const float*)d_in[9];
    const float* b_self = (const float*)d_in[10];

    const int N_SRC = in_sizes[0] / FDIM;
    const int N_DST = in_sizes[1] / FDIM;
    const int E     = in_sizes[2];

    char* ws = (char*)d_ws;
    float*    src_feat = (float*)   (ws);
    float*    dst_feat = (float*)   (ws + (size_t)N_SRC * FDIM * 4);
    float*    as_node  = (float*)   (ws + (size_t)(N_SRC + N_DST) * FDIM * 4);
    float*    ad_node  = as_node + (size_t)N_SRC * HEADS;
    unsigned* amax     = (unsigned*)(ad_node + (size_t)N_DST * HEADS);
    float*    asum     = (float*)   (amax + (size_t)N_DST * HEADS);
    float*    alpha_e  = asum + (size_t)N_DST * HEADS;

    float* out = (float*)d_out;

    // 1. init softmax accumulators
    {
        int n = N_DST * HEADS;
        init_softmax<<<(n + 255) / 256, 256, 0, stream>>>(amax, asum, n);
    }
    // 2. WMMA GEMMs (self projection writes d_out directly, incl. bias)
    gemm_x128<<<(N_SRC + 15) / 16, 256, 0, stream>>>(x_src, W_src, nullptr, src_feat, N_SRC);
    gemm_x128<<<(N_DST + 15) / 16, 256, 0, stream>>>(x_dst, W_dst, nullptr, dst_feat, N_DST);
    gemm_x128<<<(N_DST + 15) / 16, 256, 0, stream>>>(x_dst, W_self, b_self, out, N_DST);
    // 3. per-node attention dots
    node_alpha<<<(N_SRC + 7) / 8, 256, 0, stream>>>(src_feat, attS, as_node, N_SRC);
    node_alpha<<<(N_DST + 7) / 8, 256, 0, stream>>>(dst_feat, attD, ad_node, N_DST);
    // 4. edge alpha + segment max
    edge_alpha_max<<<(E + 255) / 256, 256, 0, stream>>>(esrc, edst, as_node, ad_node,
                                                        alpha_e, amax, E);
    // 5. exp + segment sum
    edge_exp_sum<<<(E + 255) / 256, 256, 0, stream>>>(edst, amax, alpha_e, asum, E);
    // 6. weighted scatter into d_out (on top of self projection)
    edge_scatter<<<(E + 7) / 8, 256, 0, stream>>>(esrc, edst, src_feat, alpha_e,
                                                  asum, out, E);
}